// RoIPoolNetwork_71373766525717
// MI455X (gfx1250) — compile-verified
//
#include <hip/hip_runtime.h>
#include <math.h>

#define B_   32
#define C_   128
#define H_   64
#define W_   64
#define NCH  2            // channel planes staged per block
#define RS   68           // LDS row stride in floats (272B -> +4 banks/row)
#define CH_STRIDE (64*RS) // per-channel LDS floats (64 rows max)

__device__ __forceinline__ int clampi(int v, int lo, int hi) {
    return v < lo ? lo : (v > hi ? hi : v);
}

#if __has_builtin(__builtin_amdgcn_global_load_async_to_lds_b128)
#define HAVE_ASYNC 1
#else
#define HAVE_ASYNC 0
#endif

// v4i matching the builtin's parameter pointee type (gcc vector_size style)
typedef int v4i __attribute__((vector_size(16)));
typedef __attribute__((address_space(1))) v4i* gptr4_t;  // global
typedef __attribute__((address_space(3))) v4i* lptr4_t;  // LDS

__device__ __forceinline__ void copy16(const float* gp, float* lp) {
#if HAVE_ASYNC
    // async DMA 16B global -> LDS, tracked by ASYNCcnt
    __builtin_amdgcn_global_load_async_to_lds_b128(
        (gptr4_t)(void*)(gp),
        (lptr4_t)(void*)(lp),
        0, 0);
#else
    *(float4*)lp = *(const float4*)gp;
#endif
}

__device__ __forceinline__ void async_wait0() {
#if HAVE_ASYNC
#if __has_builtin(__builtin_amdgcn_s_wait_asynccnt)
    __builtin_amdgcn_s_wait_asynccnt(0);
#else
    asm volatile("s_wait_asynccnt 0" ::: "memory");
#endif
#endif
}

__global__ __launch_bounds__(128)
void roi_pool_kernel(const float* __restrict__ input,
                     const float* __restrict__ lm,
                     float* __restrict__ out) {
    __shared__ __align__(16) float smem[NCH * CH_STRIDE];

    const int tid = threadIdx.x;
    const int c0  = blockIdx.x * NCH;   // channel pair
    const int lmi = blockIdx.y;         // landmark 0..7
    const int b   = blockIdx.z;

    // ---- ROI geometry (uniform within block), faithful to reference ----
    const float x0 = lm[b * 16 + 2 * lmi];
    const float y0 = lm[b * 16 + 2 * lmi + 1];
    const bool visible = (x0 > 0.0f) || (y0 > 0.0f);

    const int x1 = (int)floorf(x0 * 0.25f) - 7;
    const int y1 = (int)floorf(y0 * 0.25f) - 7;
    // torchvision slot mixup in the reference:
    const int start_w = x1;          // round(x1)
    const int start_h = x1 + 7;      // round(x2)
    // roi_w == roi_h == max(y1 - x1 + 1, 1)
    const int roi = (y1 - x1 + 1) > 1 ? (y1 - x1 + 1) : 1;
    const float binsz = (float)roi * 0.125f;

    const int h_lo  = clampi(start_h, 0, H_);          // == hs[0]
    const int h_hi  = clampi(start_h + roi, 0, H_);    // == he[7]
    const int nrows = h_hi - h_lo;

    // ---- stage window rows of NCH planes into LDS (async DMA) ----
    if (visible && nrows > 0) {
        const float* gbase = input + (size_t)(b * C_ + c0) * (H_ * W_);
        const int chunks_per_ch = nrows * 16;          // 16 x b128 per 64-float row
        const int total = NCH * chunks_per_ch;
        for (int q = tid; q < total; q += 128) {
            const int ch  = (q >= chunks_per_ch) ? 1 : 0;
            const int rem = q - ch * chunks_per_ch;
            const int row = rem >> 4;
            const int c16 = rem & 15;
            const float* gp = gbase + ch * (H_ * W_) + (h_lo + row) * W_ + c16 * 4;
            float*       lp = &smem[ch * CH_STRIDE + row * RS + c16 * 4];
            copy16(gp, lp);
        }
        async_wait0();
    }
    __syncthreads();

    // ---- per-thread bin max ----
    const int ch  = tid >> 6;       // 0..1
    const int bin = tid & 63;
    const int i   = bin >> 3;       // bin row 0..7
    const int j   = bin & 7;        // bin col 0..7

    const int hs = clampi((int)floorf((float)i * binsz)       + start_h, 0, H_);
    const int he = clampi((int)ceilf ((float)(i + 1) * binsz) + start_h, 0, H_);
    const int ws = clampi((int)floorf((float)j * binsz)       + start_w, 0, W_);
    const int we = clampi((int)ceilf ((float)(j + 1) * binsz) + start_w, 0, W_);

    float v = 0.0f;
    if (visible && (he > hs) && (we > ws)) {
        float m = -3.402823466e38f;   // finfo(float32).min
        const float* base = &smem[ch * CH_STRIDE];
        for (int r = hs - h_lo; r < he - h_lo; ++r) {
            const float* rowp = base + r * RS;
            for (int w = ws; w < we; ++w)
                m = fmaxf(m, rowp[w]);
        }
        v = m;
    }

    const int oh = 8 * (lmi >> 1) + i;
    const int ow = 8 * (lmi & 1) + j;
    out[((size_t)(b * C_ + c0 + ch) * 32 + oh) * 16 + ow] = v;
}

extern "C" void kernel_launch(void* const* d_in, const int* in_sizes, int n_in,
                              void* d_out, int out_size, void* d_ws, size_t ws_size,
                              hipStream_t stream) {
    const float* input = (const float*)d_in[0];  // [32,128,64,64] fp32
    const float* lm    = (const float*)d_in[1];  // [32,16] fp32
    float* out = (float*)d_out;                  // [32,128,32,16] fp32

    dim3 grid(C_ / NCH, 8, B_);
    roi_pool_kernel<<<grid, 128, 0, stream>>>(input, lm, out);
}